// GraphConvolution_27109833572872
// MI455X (gfx1250) — compile-verified
//
#include <hip/hip_runtime.h>

typedef float v2f __attribute__((ext_vector_type(2)));
typedef float v8f __attribute__((ext_vector_type(8)));

constexpr int kNodes  = 50000;
constexpr int kEdges  = 800000;
constexpr int kInF    = 256;
constexpr int kOutF   = 64;
constexpr int kMTiles = kNodes / 16;   // 3125 (exact)

// ---------------------------------------------------------------------------
// Kernel 1: out[n][f] = bias[f]   (fresh init every call; harness poisons once)
// ---------------------------------------------------------------------------
__global__ void __launch_bounds__(256) gcn_init_out(const float* __restrict__ bias,
                                                    float* __restrict__ out) {
  int t = blockIdx.x * 256 + threadIdx.x;
  if (t < kNodes * kOutF) out[t] = bias[t & (kOutF - 1)];
}

// ---------------------------------------------------------------------------
// Kernel 2: H = X @ W using V_WMMA_F32_16X16X4_F32 (full f32 precision).
// Block = 256 threads = 8 waves; each wave owns a full 16x64 strip of H
// (one mTile, all 4 nTiles) so every A operand is loaded from HBM exactly
// once and reused across 4 WMMAs. X traffic: 51.2 MB total (~2.5us @ 23.3TB/s).
//
// W is staged in LDS PRE-PAIRED for the WMMA B layout:
//   wlds[k2*64 + n] = { w[2*k2][n], w[2*k2+1][n] }        (64 KB)
// so each B operand is a single contiguous ds_load_b64 into an even-aligned
// VGPR pair (no re-pairing v_mov traffic in the inner loop).
//
// f32 WMMA operand layout (ISA 7.12.2):
//   A (16x4): lanes 0-15 hold row M=lane,    regs {K=0, K=1}
//             lanes 16-31 hold row M=lane-16, regs {K=2, K=3}
//   B (4x16): mirrored per column N  -> lane needs k2 = k/2 + half.
//   C/D (16x16 f32): reg r -> (M = r + 8*(lane>=16), N = lane&15)
// ---------------------------------------------------------------------------
__global__ void __launch_bounds__(256) gcn_gemm_wmma(const float* __restrict__ x,
                                                     const float* __restrict__ w,
                                                     float* __restrict__ h) {
  __shared__ v2f wlds[(kInF / 2) * kOutF];       // 64 KB of 320 KB LDS
  for (int i = threadIdx.x; i < (kInF / 2) * kOutF; i += 256) {
    int k2 = i >> 6;                             // 0..127
    int n  = i & 63;
    v2f p;
    p.x = w[(2 * k2)     * kOutF + n];           // coalesced across threads
    p.y = w[(2 * k2 + 1) * kOutF + n];
    wlds[i] = p;
  }
  __syncthreads();

  const int wave  = threadIdx.x >> 5;
  const int lane  = threadIdx.x & 31;
  const int mTile = blockIdx.x * 8 + wave;
  if (mTile >= kMTiles) return;                  // wave-uniform: EXEC stays all-1s

  const int half = lane >> 4;                    // 0: lanes 0-15, 1: lanes 16-31
  const int l16  = lane & 15;
  const int klo  = half * 2;                     // K offset {0,1} vs {2,3}

  const float* xrow = x + (size_t)(mTile * 16 + l16) * kInF + klo;
  const v2f*   wb   = wlds + half * kOutF + l16; // + (k/2)*64 per step, +16 per tile

  v8f acc0 = {}, acc1 = {}, acc2 = {}, acc3 = {};
  #pragma unroll 4
  for (int k = 0; k < kInF; k += 4) {
    v2f a;
    a.x = xrow[k];                               // single global_load_b64, 4x reuse
    a.y = xrow[k + 1];
    const int off = (k >> 1) * kOutF;            // float2 elements
    v2f b0 = wb[off +  0];                       // ds_load_b64, even-aligned pair
    v2f b1 = wb[off + 16];
    v2f b2 = wb[off + 32];
    v2f b3 = wb[off + 48];
    acc0 = __builtin_amdgcn_wmma_f32_16x16x4_f32(false, a, false, b0, (short)0, acc0, false, false);
    acc1 = __builtin_amdgcn_wmma_f32_16x16x4_f32(false, a, false, b1, (short)0, acc1, false, false);
    acc2 = __builtin_amdgcn_wmma_f32_16x16x4_f32(false, a, false, b2, (short)0, acc2, false, false);
    acc3 = __builtin_amdgcn_wmma_f32_16x16x4_f32(false, a, false, b3, (short)0, acc3, false, false);
  }

  float* hp = h + (size_t)(mTile * 16 + half * 8) * kOutF + l16;
  #pragma unroll
  for (int r = 0; r < 8; ++r) {
    hp[(size_t)r * kOutF +  0] = acc0[r];
    hp[(size_t)r * kOutF + 16] = acc1[r];
    hp[(size_t)r * kOutF + 32] = acc2[r];
    hp[(size_t)r * kOutF + 48] = acc3[r];
  }
}

// ---------------------------------------------------------------------------
// Kernel 3: out[dst] += ew[e] * H[src]  — one lane per (edge, feature-pair).
// Wave of 32 lanes covers one edge's 64 features: 256B coalesced gather of H
// (L2-resident, 12.8 MB) + hardware global_atomic_add_f32 into out (L2-resident).
// ---------------------------------------------------------------------------
__global__ void __launch_bounds__(256) gcn_scatter(const float* __restrict__ h,
                                                   const float* __restrict__ ew,
                                                   const int* __restrict__ esrc,
                                                   const int* __restrict__ edst,
                                                   float* __restrict__ out) {
  long long t = (long long)blockIdx.x * 256 + threadIdx.x;
  int e = (int)(t >> 5);
  if (e >= kEdges) return;
  int fp = ((int)t & 31) * 2;

  int   s  = esrc[e];
  int   d  = edst[e];
  float wv = ew[e];

  const float2 hv = *reinterpret_cast<const float2*>(h + (size_t)s * kOutF + fp);
  float* op = out + (size_t)d * kOutF + fp;
  unsafeAtomicAdd(op,     hv.x * wv);            // global_atomic_add_f32
  unsafeAtomicAdd(op + 1, hv.y * wv);
}

// ---------------------------------------------------------------------------
extern "C" void kernel_launch(void* const* d_in, const int* in_sizes, int n_in,
                              void* d_out, int out_size, void* d_ws, size_t ws_size,
                              hipStream_t stream) {
  const float* x    = (const float*)d_in[0];   // [50000, 256]
  const float* w    = (const float*)d_in[1];   // [256, 64]
  const float* bias = (const float*)d_in[2];   // [1, 64]
  const float* ew   = (const float*)d_in[3];   // [800000]
  const int*   esrc = (const int*)d_in[4];     // [800000]
  const int*   edst = (const int*)d_in[5];     // [800000]
  float* out = (float*)d_out;                  // [50000, 64]
  float* h   = (float*)d_ws;                   // scratch: 12.8 MB for H

  // 1) out = bias (broadcast)
  gcn_init_out<<<(kNodes * kOutF + 255) / 256, 256, 0, stream>>>(bias, out);
  // 2) H = X @ W  (WMMA f32, A loaded once per strip, B pre-paired in LDS)
  gcn_gemm_wmma<<<(kMTiles + 7) / 8, 256, 0, stream>>>(x, w, h);
  // 3) out[dst] += ew * H[src]
  gcn_scatter<<<(kEdges * 32 + 255) / 256, 256, 0, stream>>>(h, ew, esrc, edst, out);
}